// _TinyMLP_90039694393972
// MI455X (gfx1250) — compile-verified
//
#include <hip/hip_runtime.h>
#include <hip/hip_bf16.h>

#define N_GRAPHS    4096
#define D_FEAT      8
#define NUM_CLASSES 10
#define CPT         16   // nodes per thread in the segment-sum pass

typedef __attribute__((ext_vector_type(2))) float v2f;
typedef __attribute__((ext_vector_type(8))) float v8f;

// ---------------------------------------------------------------------------
// Kernel 1: zero the accumulator workspace (sums[4096*8] ++ counts[4096]).
// ---------------------------------------------------------------------------
__global__ void zero_ws_kernel(float* __restrict__ ws, int n) {
    int i = blockIdx.x * blockDim.x + threadIdx.x;
    if (i < n) ws[i] = 0.0f;
}

// ---------------------------------------------------------------------------
// Kernel 2: sorted-segment sum. Each thread walks a contiguous CPT-node chunk,
// accumulating 8 features + a count in registers; flushes with f32 global
// atomics only when the segment id changes (rare: ~977 nodes per segment).
// ---------------------------------------------------------------------------
__device__ __forceinline__ void flush_seg(float* __restrict__ sums,
                                          float* __restrict__ counts,
                                          int seg, const float* acc, float cnt) {
    const long long base = (long long)seg * D_FEAT;
#pragma unroll
    for (int f = 0; f < D_FEAT; ++f) {
        atomicAdd(&sums[base + f], acc[f]);
    }
    atomicAdd(&counts[seg], cnt);
}

__global__ void seg_sum_kernel(const float4* __restrict__ x4,   // 2 float4 per node
                               const int* __restrict__ seg_ids,
                               float* __restrict__ sums,
                               float* __restrict__ counts,
                               int n_nodes) {
    long long t     = (long long)blockIdx.x * blockDim.x + threadIdx.x;
    long long start = t * CPT;
    if (start >= n_nodes) return;
    long long end = start + CPT;
    if (end > n_nodes) end = n_nodes;

    int   cur = seg_ids[start];
    float acc[D_FEAT];
#pragma unroll
    for (int f = 0; f < D_FEAT; ++f) acc[f] = 0.0f;
    float cnt = 0.0f;

    for (long long i = start; i < end; ++i) {
        int s = seg_ids[i];
        if (s != cur) {
            flush_seg(sums, counts, cur, acc, cnt);
#pragma unroll
            for (int f = 0; f < D_FEAT; ++f) acc[f] = 0.0f;
            cnt = 0.0f;
            cur = s;
        }
        float4 lo = x4[2 * i];
        float4 hi = x4[2 * i + 1];
        acc[0] += lo.x; acc[1] += lo.y; acc[2] += lo.z; acc[3] += lo.w;
        acc[4] += hi.x; acc[5] += hi.y; acc[6] += hi.z; acc[7] += hi.w;
        cnt += 1.0f;
    }
    flush_seg(sums, counts, cur, acc, cnt);
}

// ---------------------------------------------------------------------------
// Kernel 3: pooled = sums / counts, logits = pooled @ W.T + b via
// V_WMMA_F32_16X16X4_F32. One wave (32 lanes) per 16-row tile of pooled.
//   A (16x4 f32, 2 VGPRs/lane): lanes 0-15 -> M=lane, K={k0,k0+1};
//                               lanes 16-31 -> M=lane-16, K={k0+2,k0+3}.
//   B (4x16 f32, 2 VGPRs/lane): lanes 0-15 -> N=lane, K={k0,k0+1};
//                               lanes 16-31 -> N=lane-16, K={k0+2,k0+3}.
//   C/D (16x16 f32, 8 VGPRs):   VGPR v, lanes 0-15 -> M=v,   N=lane;
//                               lanes 16-31          -> M=v+8, N=lane-16.
// K=8 handled as two chained K=4 WMMA steps. Columns 10..15 are zero-padded
// via clamped loads * mask (no divergence: EXEC stays all-ones for WMMA).
// ---------------------------------------------------------------------------
__global__ void pool_gemm_wmma_kernel(const float* __restrict__ sums,
                                      const float* __restrict__ counts,
                                      const float* __restrict__ W,      // [10][8] row-major
                                      const float* __restrict__ bias,   // [10]
                                      float* __restrict__ out) {        // [4096][10]
    const int lane = threadIdx.x;        // 0..31
    const int tile = blockIdx.x;         // 0..255
    const int row0 = tile * 16;
    const int half = lane >> 4;          // 0 or 1
    const int mn   = lane & 15;          // M (for A) / N (for B,C,D) within tile

    // ---- A operand: pooled rows (divide-on-the-fly) ----
    const int   r   = row0 + mn;
    const float cv  = counts[r];
    const float inv = (cv > 0.0f) ? (1.0f / cv) : 0.0f;
    const float* srow = sums + (long long)r * D_FEAT;
    const int c0 = half * 2;             // K offset within a K=4 step
    v2f a0, a1;
    a0.x = srow[c0 + 0] * inv;  a0.y = srow[c0 + 1] * inv;   // K-step 0: k = 0..3
    a1.x = srow[c0 + 4] * inv;  a1.y = srow[c0 + 5] * inv;   // K-step 1: k = 4..7

    // ---- B operand: W^T tile, zero-padded for N >= 10 (clamped, branch-free) ----
    const float nmask = (mn < NUM_CLASSES) ? 1.0f : 0.0f;
    const int   nclmp = (mn < NUM_CLASSES) ? mn : (NUM_CLASSES - 1);
    const float* wrow = W + nclmp * D_FEAT;   // B[k][n] = W[n][k]
    v2f b0, b1;
    b0.x = wrow[c0 + 0] * nmask;  b0.y = wrow[c0 + 1] * nmask;
    b1.x = wrow[c0 + 4] * nmask;  b1.y = wrow[c0 + 5] * nmask;

    // ---- C operand: broadcast bias ----
    const float cinit = bias[nclmp] * nmask;
    v8f c = {cinit, cinit, cinit, cinit, cinit, cinit, cinit, cinit};

    // ---- two K=4 steps: D = A1*B1 + (A0*B0 + C) ----
    c = __builtin_amdgcn_wmma_f32_16x16x4_f32(false, a0, false, b0,
                                              (short)0, c, false, false);
    c = __builtin_amdgcn_wmma_f32_16x16x4_f32(false, a1, false, b1,
                                              (short)0, c, false, false);

    // ---- store the 10 valid columns ----
    if (mn < NUM_CLASSES) {
#pragma unroll
        for (int v = 0; v < 8; ++v) {
            const int m = row0 + v + half * 8;
            out[m * NUM_CLASSES + mn] = c[v];
        }
    }
}

// ---------------------------------------------------------------------------
// Launch wrapper
// ---------------------------------------------------------------------------
extern "C" void kernel_launch(void* const* d_in, const int* in_sizes, int n_in,
                              void* d_out, int out_size, void* d_ws, size_t ws_size,
                              hipStream_t stream) {
    const float* x       = (const float*)d_in[0];   // [N_NODES, 8]
    const int*   seg_ids = (const int*)d_in[1];     // [N_NODES] sorted
    // d_in[2] input_ids, d_in[3] attention_mask: unused by the reference forward
    const float* W       = (const float*)d_in[4];   // [10, 8]
    const float* bias    = (const float*)d_in[5];   // [10]
    float*       out     = (float*)d_out;           // [4096, 10]

    float* sums   = (float*)d_ws;                   // [4096, 8]
    float* counts = sums + N_GRAPHS * D_FEAT;       // [4096]

    const int n_nodes  = in_sizes[1];
    const int ws_elems = N_GRAPHS * (D_FEAT + 1);

    zero_ws_kernel<<<(ws_elems + 255) / 256, 256, 0, stream>>>(sums, ws_elems);

    const long long nthreads = ((long long)n_nodes + CPT - 1) / CPT;
    const int       nblocks  = (int)((nthreads + 255) / 256);
    seg_sum_kernel<<<nblocks, 256, 0, stream>>>((const float4*)x, seg_ids,
                                                sums, counts, n_nodes);

    pool_gemm_wmma_kernel<<<N_GRAPHS / 16, 32, 0, stream>>>(sums, counts, W, bias, out);
}